// PowerVector_56667798504193
// MI455X (gfx1250) — compile-verified
//
#include <hip/hip_runtime.h>
#include <cstdint>
#include <cstddef>

typedef __attribute__((ext_vector_type(2))) float v2f;
typedef __attribute__((ext_vector_type(8))) float v8f;

#define NB    4
#define NT    1024
#define NCH   4
#define NF    481
#define NK    48
#define NCOVC 16
#define NPV   16
#define NCHUNKP 122  /* padded chunk count: 121 real (ceil(481/4)) + 1 zero */
#define NHALF  61    /* chunks per wave-half: 2*61 = 122 */
#define NTILE  3     /* 48 bands / 16 */
#define NFPAD  (NF + 7) /* adj rows incl. zero pad covering f = 481..487 */
#define DT_MS 10.0f

// ---------------------------------------------------------------------------
// Kernel 0: pack the complex band matrix (F,K) into WMMA A-fragments.
// A-matrix 16x4 f32 layout (ISA 7.12.2): lane = m + 16*(k/2), vgpr j = k%2.
// Stored as float2 per lane so kernel 1 loads one b64 per fragment.
// Chunk 121 (f >= 484) is all zeros so both wave-halves run 61 chunks.
// ---------------------------------------------------------------------------
__global__ void pack_band_kernel(const float* __restrict__ band_r,
                                 const float* __restrict__ band_i,
                                 float2* __restrict__ Ar,
                                 float2* __restrict__ Ai) {
  int idx = blockIdx.x * blockDim.x + threadIdx.x;
  const int total = NTILE * NCHUNKP * 32;
  if (idx >= total) return;
  int lane  = idx & 31;
  int cidx  = idx >> 5;
  int tile  = cidx / NCHUNKP;
  int chunk = cidx % NCHUNKP;
  int m  = lane & 15;          // band index within tile (M)
  int h  = lane >> 4;          // K half
  int kb = tile * 16 + m;      // global band column
  int f0 = chunk * 4 + 2 * h;  // frequency for vgpr0; vgpr1 = f0+1
  float2 r, im;
  r.x  = (f0     < NF) ? band_r[(size_t)f0       * NK + kb] : 0.f;
  r.y  = (f0 + 1 < NF) ? band_r[(size_t)(f0 + 1) * NK + kb] : 0.f;
  im.x = (f0     < NF) ? band_i[(size_t)f0       * NK + kb] : 0.f;
  im.y = (f0 + 1 < NF) ? band_i[(size_t)(f0 + 1) * NK + kb] : 0.f;
  Ar[idx] = r;
  Ai[idx] = im;
}

// ---------------------------------------------------------------------------
// Kernel 1: per-frame pipeline. One workgroup = one (b,t); 6 waves = 3 band
// tiles x 2 F-range halves (constant 61 chunks each -> uniform trip count,
// no exec masking around WMMA). Build phase-adjusted covariances in LDS
// (Hermitian: only 10 of 16 components computed), then complex GEMM via
// V_WMMA_F32_16X16X4_F32 with 4 independent accumulators per wave.
// ---------------------------------------------------------------------------
__launch_bounds__(192)
__global__ void frame_gemm_kernel(const float* __restrict__ bins_r,
                                  const float* __restrict__ bins_i,
                                  const float2* __restrict__ Ar,
                                  const float2* __restrict__ Ai,
                                  float2* __restrict__ bc) {
  __shared__ float s_xr[NCH * NF], s_xi[NCH * NF];
  __shared__ float s_pw[NF];
  __shared__ float s_adjre[NFPAD * NCOVC];  // zero rows pad the last chunks
  __shared__ float s_adjim[NFPAD * NCOVC];
  __shared__ float s_bcre[NK * NCOVC], s_bcim[NK * NCOVC];
  __shared__ float s_ds[NK];

  const int tid = threadIdx.x;
  const int b = blockIdx.x / NT;
  const int t = blockIdx.x % NT;

  // Stage STFT bins for this frame (4 ch x 481 f, complex)
  const float* br = bins_r + (size_t)(b * NT + t) * NCH * NF;
  const float* bi = bins_i + (size_t)(b * NT + t) * NCH * NF;
  for (int e = tid; e < NCH * NF; e += 192) { s_xr[e] = br[e]; s_xi[e] = bi[e]; }
  // Zero the pad rows (f = 481..487)
  for (int e = tid; e < (NFPAD - NF) * NCOVC; e += 192) {
    s_adjre[NF * NCOVC + e] = 0.f;
    s_adjim[NF * NCOVC + e] = 0.f;
  }
  __syncthreads();

  // Per-bin power = trace of raw covariance
  for (int f = tid; f < NF; f += 192) {
    float p = 0.f;
    for (int ch = 0; ch < NCH; ++ch) {
      float xr = s_xr[ch * NF + f], xi = s_xi[ch * NF + f];
      p += xr * xr + xi * xi;
    }
    s_pw[f] = fmaxf(p, 1e-20f);
  }
  __syncthreads();

  // Phase-adjusted normalized covariance (Hermitian: compute 4 diag + 6
  // upper-triangle components, mirror conjugates):
  //   adj[f,ij] = |cov_n[f,ij]| * unit(conj(cov[fa,ij]) * cov[fb,ij])
  // angle is invariant to the (positive) power normalization.
  {
    const int PI[10] = {0, 1, 2, 3, 0, 0, 0, 1, 1, 2};
    const int PJ[10] = {0, 1, 2, 3, 1, 2, 3, 2, 3, 3};
    for (int e = tid; e < NF * 10; e += 192) {
      int f = e / 10, p = e % 10;
      int i = PI[p], j = PJ[p];
      float ar = s_xr[i * NF + f], ai = s_xi[i * NF + f];
      float br0 = s_xr[j * NF + f], bi0 = s_xi[j * NF + f];
      // cov = x_i * conj(x_j)
      float cre = ar * br0 + ai * bi0;
      float cim = ai * br0 - ar * bi0;
      float invp = 1.f / s_pw[f];
      if (p < 4) {
        s_adjre[f * NCOVC + p * 5] = cre * invp;  // diagonal kept as-is (real)
        s_adjim[f * NCOVC + p * 5] = 0.f;
      } else {
        int fa = (f == 0) ? 0 : ((f == NF - 1) ? NF - 3 : f - 1);
        int fb = (f == 0) ? 2 : ((f == NF - 1) ? NF - 1 : f + 1);
        float a1r = s_xr[i * NF + fa], a1i = s_xi[i * NF + fa];
        float b1r = s_xr[j * NF + fa], b1i = s_xi[j * NF + fa];
        float c1re = a1r * b1r + a1i * b1i;
        float c1im = a1i * b1r - a1r * b1i;
        float a2r = s_xr[i * NF + fb], a2i = s_xi[i * NF + fb];
        float b2r = s_xr[j * NF + fb], b2i = s_xi[j * NF + fb];
        float c2re = a2r * b2r + a2i * b2i;
        float c2im = a2i * b2r - a2r * b2i;
        // z = conj(c1) * c2
        float zre = c1re * c2re + c1im * c2im;
        float zim = c1re * c2im - c1im * c2re;
        float zl = sqrtf(zre * zre + zim * zim);
        float ure, uim;
        if (zl > 0.f) { float iz = 1.f / zl; ure = zre * iz; uim = zim * iz; }
        else          { ure = 1.f; uim = 0.f; }        // angle(0) = 0
        float mag = sqrtf(cre * cre + cim * cim) * invp;
        float re = mag * ure, im = mag * uim;
        s_adjre[f * NCOVC + i * 4 + j] = re;
        s_adjim[f * NCOVC + i * 4 + j] = im;
        s_adjre[f * NCOVC + j * 4 + i] = re;      // Hermitian mirror
        s_adjim[f * NCOVC + j * 4 + i] = -im;
      }
    }
  }
  __syncthreads();

  // Complex GEMM: out[16 bands x 16 cov] += band_tile(16x4) x adj_chunk(4x16)
  // per 4-frequency chunk. 4 independent f32 accumulators per wave (no WMMA
  // RAW chains within a chunk); each tile covered by 2 waves x 61 chunks.
  const int wave = tid >> 5;
  const int tile = wave >> 1;      // band tile 0..2
  const int half = wave & 1;       // F-range half
  const int lane = tid & 31;
  const int n = lane & 15;         // cov component (B/N column)
  const int h = lane >> 4;         // K half
  const int cbeg = half * NHALF;   // 0 or 61; both halves run exactly 61
  v8f acc_rr = {}; v8f acc_ii = {}; v8f acc_ri = {}; v8f acc_ir = {};
  const float2* ArT = Ar + ((size_t)tile * NCHUNKP + cbeg) * 32;
  const float2* AiT = Ai + ((size_t)tile * NCHUNKP + cbeg) * 32;
  #pragma unroll 4
  for (int c = 0; c < NHALF; ++c) {
    float2 arl = ArT[c * 32 + lane];
    float2 ail = AiT[c * 32 + lane];
    v2f a_r; a_r.x = arl.x; a_r.y = arl.y;
    v2f a_i; a_i.x = ail.x; a_i.y = ail.y;
    int f2 = (cbeg + c) * 4 + 2 * h; // B 4x16 layout mirrors A: lane=n+16*(k/2)
    v2f b_r, b_i;
    b_r.x = s_adjre[f2 * NCOVC + n];
    b_r.y = s_adjre[(f2 + 1) * NCOVC + n];
    b_i.x = s_adjim[f2 * NCOVC + n];
    b_i.y = s_adjim[(f2 + 1) * NCOVC + n];
    acc_rr = __builtin_amdgcn_wmma_f32_16x16x4_f32(false, a_r, false, b_r,
                                                   (short)0, acc_rr, false, false);
    acc_ii = __builtin_amdgcn_wmma_f32_16x16x4_f32(false, a_i, false, b_i,
                                                   (short)0, acc_ii, false, false);
    acc_ri = __builtin_amdgcn_wmma_f32_16x16x4_f32(false, a_r, false, b_i,
                                                   (short)0, acc_ri, false, false);
    acc_ir = __builtin_amdgcn_wmma_f32_16x16x4_f32(false, a_i, false, b_r,
                                                   (short)0, acc_ir, false, false);
  }
  // C/D layout: vgpr v -> row (v + 8*h), col n. Re = rr - ii, Im = ri + ir.
  // Half 0 writes its partials, half 1 accumulates on top after a barrier.
  if (half == 0) {
    #pragma unroll
    for (int v = 0; v < 8; ++v) {
      int kb = tile * 16 + v + 8 * h;
      s_bcre[kb * NCOVC + n] = acc_rr[v] - acc_ii[v];
      s_bcim[kb * NCOVC + n] = acc_ri[v] + acc_ir[v];
    }
  }
  __syncthreads();
  if (half == 1) {
    #pragma unroll
    for (int v = 0; v < 8; ++v) {
      int kb = tile * 16 + v + 8 * h;
      s_bcre[kb * NCOVC + n] += acc_rr[v] - acc_ii[v];
      s_bcim[kb * NCOVC + n] += acc_ri[v] + acc_ir[v];
    }
  }
  __syncthreads();

  // Per-band normalization by real diagonal sum
  if (tid < NK) {
    float d = s_bcre[tid * NCOVC + 0] + s_bcre[tid * NCOVC + 5] +
              s_bcre[tid * NCOVC + 10] + s_bcre[tid * NCOVC + 15];
    s_ds[tid] = fmaxf(d, 1e-20f);
  }
  __syncthreads();

  float2* bco = bc + (size_t)(b * NT + t) * NK * NCOVC;
  for (int e = tid; e < NK * NCOVC; e += 192) {
    int k = e / NCOVC;
    float inv = 1.f / s_ds[k];
    float2 o; o.x = s_bcre[e] * inv; o.y = s_bcim[e] * inv;
    bco[e] = o;
  }
}

// ---------------------------------------------------------------------------
// Kernel 2: IIR over frames, in place. One thread per (b,k,c) complex
// sequence; t=0 passes through (matches the reference scan init).
// ---------------------------------------------------------------------------
__global__ void iir_kernel(float2* __restrict__ bc, const float* __restrict__ tau) {
  int id = blockIdx.x * blockDim.x + threadIdx.x;
  if (id >= NB * NK * NCOVC) return;
  int b = id / (NK * NCOVC);
  int r = id % (NK * NCOVC);
  int k = r / NCOVC;
  float alpha = expf(-DT_MS / tau[k]);
  float om = 1.f - alpha;
  size_t base = (size_t)b * NT * NK * NCOVC + r;
  float2 y = bc[base];
  for (int t = 1; t < NT; ++t) {
    size_t idx = base + (size_t)t * NK * NCOVC;
    float2 x = bc[idx];
    y.x = alpha * y.x + om * x.x;
    y.y = alpha * y.y + om * x.y;
    bc[idx] = y;
  }
}

// ---------------------------------------------------------------------------
// Kernel 3: pv[b,k,t,p] = Re( c2pv[p,:] . bc[b,t,k,:] ). One thread per
// (b,k,t): 128B contiguous read, 64B contiguous write.
// ---------------------------------------------------------------------------
__launch_bounds__(256)
__global__ void pv_kernel(const float2* __restrict__ bc,
                          const float* __restrict__ c2r,
                          const float* __restrict__ c2i,
                          float* __restrict__ out) {
  __shared__ float s_cr[NPV * NCOVC], s_ci[NPV * NCOVC];
  for (int e = threadIdx.x; e < NPV * NCOVC; e += blockDim.x) {
    s_cr[e] = c2r[e]; s_ci[e] = c2i[e];
  }
  __syncthreads();
  int id = blockIdx.x * blockDim.x + threadIdx.x;
  if (id >= NB * NK * NT) return;
  int b = id / (NK * NT);
  int rem = id % (NK * NT);
  int k = rem / NT;
  int t = rem % NT;
  const float2* y = bc + ((size_t)(b * NT + t) * NK + k) * NCOVC;
  float yre[NCOVC], yim[NCOVC];
  #pragma unroll
  for (int c = 0; c < NCOVC; ++c) { float2 v = y[c]; yre[c] = v.x; yim[c] = v.y; }
  float* o = out + ((size_t)(b * NK + k) * NT + t) * NPV;
  #pragma unroll
  for (int p = 0; p < NPV; ++p) {
    float acc = 0.f;
    #pragma unroll
    for (int c = 0; c < NCOVC; ++c)
      acc += s_cr[p * NCOVC + c] * yre[c] - s_ci[p * NCOVC + c] * yim[c];
    o[p] = acc;
  }
}

// ---------------------------------------------------------------------------
extern "C" void kernel_launch(void* const* d_in, const int* in_sizes, int n_in,
                              void* d_out, int out_size, void* d_ws, size_t ws_size,
                              hipStream_t stream) {
  const float* bins_r = (const float*)d_in[0];
  const float* bins_i = (const float*)d_in[1];
  const float* band_r = (const float*)d_in[2];
  const float* band_i = (const float*)d_in[3];
  const float* c2r    = (const float*)d_in[4];
  const float* c2i    = (const float*)d_in[5];
  const float* tau    = (const float*)d_in[6];

  // Workspace: packed A fragments (2 x ~94 KB) + bc intermediate (~25.2 MB)
  const size_t apack = (size_t)NTILE * NCHUNKP * 32;
  float2* Ar = (float2*)d_ws;
  float2* Ai = Ar + apack;
  float2* bc = Ai + apack;

  pack_band_kernel<<<(int)((apack + 255) / 256), 256, 0, stream>>>(band_r, band_i, Ar, Ai);
  frame_gemm_kernel<<<NB * NT, 192, 0, stream>>>(bins_r, bins_i, Ar, Ai, bc);
  iir_kernel<<<(NB * NK * NCOVC + 255) / 256, 256, 0, stream>>>(bc, tau);
  pv_kernel<<<(NB * NK * NT + 255) / 256, 256, 0, stream>>>(bc, c2r, c2i, (float*)d_out);
}